// CrossProdMean_16071767621818
// MI455X (gfx1250) — compile-verified
//
#include <hip/hip_runtime.h>

#define DIM 1024
#define HID 1024
#define NROWS (8 * 4096)

typedef __attribute__((ext_vector_type(2))) float v2f;
typedef __attribute__((ext_vector_type(8))) float v8f;

__device__ __forceinline__ float wave_sum(float s) {
#pragma unroll
  for (int off = 16; off > 0; off >>= 1) s += __shfl_xor(s, off, 32);
  return s;
}

// Kernel 1: wbar[d] = mean_h Wb[d,h]; bbmean = mean(bb)
__global__ __launch_bounds__(256) void prep_kernel(const float* __restrict__ Wb,
                                                   const float* __restrict__ bb,
                                                   float* __restrict__ wbar,
                                                   float* __restrict__ bbmean) {
  const int wave = threadIdx.x >> 5;
  const int lane = threadIdx.x & 31;
  if (blockIdx.x < 128) {
    const int row = blockIdx.x * 8 + wave;  // 128*8 = 1024 rows
    const float4* wr = (const float4*)(Wb + (size_t)row * HID);
    float s = 0.f;
#pragma unroll
    for (int i = 0; i < 8; ++i) {
      float4 v = wr[lane + 32 * i];
      s += v.x + v.y + v.z + v.w;
    }
    s = wave_sum(s);
    if (lane == 0) wbar[row] = s * (1.0f / HID);
  } else if (wave == 0) {
    const float4* br = (const float4*)bb;
    float s = 0.f;
#pragma unroll
    for (int i = 0; i < 8; ++i) {
      float4 v = br[lane + 32 * i];
      s += v.x + v.y + v.z + v.w;
    }
    s = wave_sum(s);
    if (lane == 0) *bbmean = s * (1.0f / HID);
  }
}

// Kernel 2: r[n] = dot(x[n,:], wbar) + bbmean   (wave per row)
__global__ __launch_bounds__(256) void rowdot_kernel(const float* __restrict__ x,
                                                     const float* __restrict__ wbar,
                                                     const float* __restrict__ bbmean,
                                                     float* __restrict__ r) {
  const int wave = threadIdx.x >> 5;
  const int lane = threadIdx.x & 31;
  const int n = blockIdx.x * 8 + wave;
  const float4* xr = (const float4*)(x + (size_t)n * DIM);
  const float4* wr = (const float4*)wbar;
  float s = 0.f;
#pragma unroll
  for (int i = 0; i < 8; ++i) {
    float4 a = xr[lane + 32 * i];
    float4 b = wr[lane + 32 * i];
    s += a.x * b.x + a.y * b.y + a.z * b.z + a.w * b.w;
  }
  s = wave_sum(s);
  if (lane == 0) r[n] = s + bbmean[0];
}

// Kernel 3: out[n,h] = g[h] * (x@Wa + ba)[n,h] * r[n]
// Block = 16 waves (512 thr); block tile M=128 x N=128; K chunk = 32 in LDS.
// Waves in 4x4 grid; each wave owns a 32x32 C patch = 2x2 WMMA 16x16 tiles.
// Both A and B fragments are single aligned float2 LDS reads (no repack movs).
__global__ __launch_bounds__(512) void fused_gemm_kernel(const float* __restrict__ x,
                                                         const float* __restrict__ Wa,
                                                         const float* __restrict__ ba,
                                                         const float* __restrict__ g,
                                                         const float* __restrict__ r,
                                                         float* __restrict__ out) {
  // xs stride 36: {36m mod 64} is 16 distinct banks (= 0 mod 4 set); lanes 16-31
  // read col+2 -> disjoint = 2 mod 4 set -> conflict-free 32-lane b64 reads.
  __shared__ float xs[128][36];    // 128 rows (M) x 32 k      (18 KB)
  // K-pair interleaved Wa tile: wasI[k/2][2*col + (k&1)].
  // Pair-row stride 288 floats = 1152 B = 32 banks mod 64 -> lanes 0-15 (pair kk/2)
  // and lanes 16-31 (pair kk/2+1) hit disjoint 32-bank halves.
  __shared__ float wasI[16][288];  // 16 k-pairs x 128 cols x 2 (+32 pad)  (18 KB)

  const int n_blk = blockIdx.x * 128;
  const int m_blk = blockIdx.y * 128;
  const int t = threadIdx.x;
  const int wave = t >> 5;
  const int lane = t & 31;
  const int half = lane >> 4;    // 0: lanes 0-15, 1: lanes 16-31
  const int lm = lane & 15;
  const int ksel = half << 1;    // K offset 0 or 2 per f32 A/B fragment layout
  const int wm = (wave >> 2) * 32;  // wave's M base within block tile (0..96)
  const int wn = (wave & 3) * 32;   // wave's N base within block tile (0..96)

  v8f acc[2][2] = {};

  // staging coordinates
  const int xr_row = t >> 3;          // x tile: 128 rows x 8 float4
  const int xr_q = (t & 7) << 2;
  const int wp = t >> 5;              // Wa tile: 16 pair-rows x 32 col-groups
  const int wq = (t & 31) << 2;

  for (int k0 = 0; k0 < DIM; k0 += 32) {
    // stage x tile: 128x32 f32 = 1024 float4, two per thread
#pragma unroll
    for (int s = 0; s < 2; ++s) {
      const int row = xr_row + s * 64;
      float4 v = *(const float4*)(x + (size_t)(m_blk + row) * DIM + k0 + xr_q);
      *(float4*)(&xs[row][xr_q]) = v;
    }
    // stage Wa tile K-pair interleaved: rows 2*wp, 2*wp+1, cols wq..wq+3
    {
      float4 w0 = *(const float4*)(Wa + (size_t)(k0 + 2 * wp) * HID + n_blk + wq);
      float4 w1 = *(const float4*)(Wa + (size_t)(k0 + 2 * wp + 1) * HID + n_blk + wq);
      float4 lo = {w0.x, w1.x, w0.y, w1.y};
      float4 hi = {w0.z, w1.z, w0.w, w1.w};
      *(float4*)(&wasI[wp][wq * 2]) = lo;
      *(float4*)(&wasI[wp][wq * 2 + 4]) = hi;
    }
    __syncthreads();

#pragma unroll
    for (int kk = 0; kk < 32; kk += 4) {
      v2f a[2], b[2];
#pragma unroll
      for (int i = 0; i < 2; ++i) {
        // A (16x4): lanes 0-15 -> K=kk,kk+1 ; lanes 16-31 -> K=kk+2,kk+3 ; M=lm
        float2 av = *(const float2*)(&xs[wm + i * 16 + lm][kk + ksel]);
        a[i].x = av.x;
        a[i].y = av.y;
      }
#pragma unroll
      for (int j = 0; j < 2; ++j) {
        // B (4x16): K pair (kk+ksel, kk+ksel+1) contiguous in wasI ; N = lm
        float2 bv = *(const float2*)(&wasI[(kk >> 1) + half][(wn + j * 16 + lm) * 2]);
        b[j].x = bv.x;
        b[j].y = bv.y;
      }
#pragma unroll
      for (int i = 0; i < 2; ++i)
#pragma unroll
        for (int j = 0; j < 2; ++j)
          acc[i][j] = __builtin_amdgcn_wmma_f32_16x16x4_f32(
              false, a[i], false, b[j], (short)0, acc[i][j], false, false);
    }
    __syncthreads();
  }

  // Epilogue: C VGPR jj -> row jj (lanes 0-15) / jj+8 (lanes 16-31), col = lane%16.
  // Non-temporal stores: keep the write-once output out of L2 so x/Wa stay resident.
#pragma unroll
  for (int j = 0; j < 2; ++j) {
    const int col = n_blk + wn + j * 16 + lm;
    const float gv = g[col];
    const float bav = ba[col];
#pragma unroll
    for (int i = 0; i < 2; ++i) {
      const int rbase = m_blk + wm + i * 16 + half * 8;
#pragma unroll
      for (int jj = 0; jj < 8; ++jj) {
        const int row = rbase + jj;
        const float val = gv * (acc[i][j][jj] + bav) * r[row];
        __builtin_nontemporal_store(val, out + (size_t)row * HID + col);
      }
    }
  }
}

extern "C" void kernel_launch(void* const* d_in, const int* in_sizes, int n_in,
                              void* d_out, int out_size, void* d_ws, size_t ws_size,
                              hipStream_t stream) {
  const float* x = (const float*)d_in[0];
  const float* Wa = (const float*)d_in[1];
  const float* ba = (const float*)d_in[2];
  const float* Wb = (const float*)d_in[3];
  const float* bb = (const float*)d_in[4];
  const float* g = (const float*)d_in[5];
  float* out = (float*)d_out;

  float* ws = (float*)d_ws;
  float* wbar = ws;        // 1024 floats
  float* bbm = ws + 1024;  // 1 float
  float* r = ws + 2048;    // 32768 floats

  prep_kernel<<<129, 256, 0, stream>>>(Wb, bb, wbar, bbm);
  rowdot_kernel<<<NROWS / 8, 256, 0, stream>>>(x, wbar, bbm, r);
  dim3 grid(HID / 128, NROWS / 128);
  fused_gemm_kernel<<<grid, 512, 0, stream>>>(x, Wa, ba, g, r, out);
}